// Encoder_2353642078838
// MI455X (gfx1250) — compile-verified
//
#include <hip/hip_runtime.h>

typedef __attribute__((ext_vector_type(2))) float v2f;
typedef __attribute__((ext_vector_type(8))) float v8f;

#define HAVE_WMMA_F32 __has_builtin(__builtin_amdgcn_wmma_f32_16x16x4_f32)

// ---------------------------------------------------------------------------
// Degree / normalization kernels
// ---------------------------------------------------------------------------
__global__ __launch_bounds__(256) void zero_f32(float* __restrict__ p, int n) {
    int i = blockIdx.x * 256 + threadIdx.x;
    if (i < n) p[i] = 0.0f;
}

__global__ __launch_bounds__(256) void count_deg(const int* __restrict__ dst,
                                                 float* __restrict__ deg, int E) {
    int i = blockIdx.x * 256 + threadIdx.x;
    if (i < E) atomicAdd(&deg[dst[i]], 1.0f);
}

__global__ __launch_bounds__(256) void make_dinv(float* __restrict__ deg_dinv, int n) {
    int i = blockIdx.x * 256 + threadIdx.x;
    if (i < n) deg_dinv[i] = rsqrtf(deg_dinv[i] + 1.0f);  // +1 for self-loop
}

// ---------------------------------------------------------------------------
// GEMM via V_WMMA_F32_16X16X4_F32, fused epilogue:
//   H[r][c]   = (RELU_IN ? relu(X) : X) @ W          (raw, for edge gather)
//   AGG[r][c] = H[r][c] * dinv[r]^2 + bias[c]        (self-loop + bias init)
// Block = 256 threads = 8 waves. Each wave owns one 16x16 D tile.
//   NCOLS=128 -> 1 row-tile x 8 col-tiles (16 rows/block)
//   NCOLS=64  -> 2 row-tiles x 4 col-tiles (32 rows/block)
// ---------------------------------------------------------------------------
template <int NCOLS, bool RELU_IN>
__global__ __launch_bounds__(256) void gcn_gemm(const float* __restrict__ X,
                                                const float* __restrict__ W,
                                                const float* __restrict__ bias,
                                                const float* __restrict__ dinv,
                                                float* __restrict__ H,
                                                float* __restrict__ AGG, int N) {
    constexpr int K    = 128;
    constexpr int COLT = NCOLS / 16;  // 8 or 4
    constexpr int ROWT = 8 / COLT;    // 1 or 2
    constexpr int ROWS = ROWT * 16;   // 16 or 32
    constexpr int LDK  = K + 4;       // pad: kills 16-way LDS bank conflicts

    __shared__ float sX[ROWS * LDK];

    const int tid  = threadIdx.x;
    const int lane = tid & 31;
    const int wave = tid >> 5;
    const int row0 = blockIdx.x * ROWS;

    // Cooperative load of X tile (ROWS x 128) into LDS, optional fused ReLU.
    for (int idx = tid * 4; idx < ROWS * K; idx += 256 * 4) {
        int r = idx >> 7;        // / 128
        int c = idx & (K - 1);   // % 128
        int gr = row0 + r;
        float4 v = make_float4(0.0f, 0.0f, 0.0f, 0.0f);
        if (gr < N) v = *(const float4*)(X + (size_t)gr * K + c);
        if (RELU_IN) {
            v.x = fmaxf(v.x, 0.0f); v.y = fmaxf(v.y, 0.0f);
            v.z = fmaxf(v.z, 0.0f); v.w = fmaxf(v.w, 0.0f);
        }
        float* p = &sX[r * LDK + c];
        p[0] = v.x; p[1] = v.y; p[2] = v.z; p[3] = v.w;
    }
    __syncthreads();

    const int rt   = wave / COLT;
    const int ct   = wave % COLT;
    const int col0 = ct * 16;
    const int m    = lane & 15;  // A row / B,D column within tile
    const int kh   = lane >> 4;  // selects K sub-pair (ISA 16x4 f32 layout)

    v8f acc = {0.f, 0.f, 0.f, 0.f, 0.f, 0.f, 0.f, 0.f};

    const float* wcol = W + col0 + m;       // our column of W
    const float* xrow = &sX[(rt * 16 + m) * LDK];

#if HAVE_WMMA_F32
#pragma unroll
    for (int kk = 0; kk < K; kk += 4) {
        const int kb = kk + 2 * kh;
        v2f a, b;
        a.x = xrow[kb];
        a.y = xrow[kb + 1];
        b.x = wcol[(size_t)(kb + 0) * NCOLS];
        b.y = wcol[(size_t)(kb + 1) * NCOLS];
        // D = A(16x4,f32) x B(4x16,f32) + C ; 8-arg form
        acc = __builtin_amdgcn_wmma_f32_16x16x4_f32(
            /*neg_a=*/false, a, /*neg_b=*/false, b,
            /*c_mod=*/(short)0, acc, /*reuse_a=*/false, /*reuse_b=*/false);
    }
#else
    // Scalar fallback (host pass / builtin missing): same lane->element map.
#pragma unroll 4
    for (int kk = 0; kk < K; ++kk) {
        float wv = wcol[(size_t)kk * NCOLS];
#pragma unroll
        for (int v = 0; v < 8; ++v)
            acc[v] += sX[(rt * 16 + kh * 8 + v) * LDK + kk] * wv;
    }
#endif

    // Epilogue: D layout -> lane covers column m, rows kh*8 + v.
    const float bcol = bias[col0 + m];
#pragma unroll
    for (int v = 0; v < 8; ++v) {
        int r = row0 + rt * 16 + kh * 8 + v;
        if (r < N) {
            float hval = acc[v];
            size_t off = (size_t)r * NCOLS + col0 + m;
            H[off] = hval;
            float d = dinv[r];
            AGG[off] = hval * d * d + bcol;
        }
    }
}

// ---------------------------------------------------------------------------
// Edge scatter: AGG[dst] += H[src] * (dinv[src]*dinv[dst]).  One wave/edge,
// each lane handles C/32 contiguous channels; f32 atomics hit L2 atomic units.
// ---------------------------------------------------------------------------
template <int C>
__global__ __launch_bounds__(256) void gcn_scatter(const float* __restrict__ H,
                                                   const float* __restrict__ dinv,
                                                   const int* __restrict__ src,
                                                   const int* __restrict__ dst,
                                                   float* __restrict__ AGG, int E) {
    int wid  = blockIdx.x * 8 + (threadIdx.x >> 5);
    int lane = threadIdx.x & 31;
    if (wid >= E) return;
    int s = src[wid];
    int d = dst[wid];
    float coeff = dinv[s] * dinv[d];
    constexpr int V = C / 32;  // 4 or 2
    const float* hp = H + (size_t)s * C + lane * V;
    float* ap = AGG + (size_t)d * C + lane * V;
    if constexpr (V == 4) {
        float4 hv = *(const float4*)hp;
        atomicAdd(ap + 0, hv.x * coeff);
        atomicAdd(ap + 1, hv.y * coeff);
        atomicAdd(ap + 2, hv.z * coeff);
        atomicAdd(ap + 3, hv.w * coeff);
    } else {
        float2 hv = *(const float2*)hp;
        atomicAdd(ap + 0, hv.x * coeff);
        atomicAdd(ap + 1, hv.y * coeff);
    }
}

// ---------------------------------------------------------------------------
extern "C" void kernel_launch(void* const* d_in, const int* in_sizes, int n_in,
                              void* d_out, int out_size, void* d_ws, size_t ws_size,
                              hipStream_t stream) {
    const float* x  = (const float*)d_in[0];
    const int*   ei = (const int*)d_in[1];   // [2, E] flat
    const float* W1 = (const float*)d_in[2];
    const float* b1 = (const float*)d_in[3];
    const float* W2 = (const float*)d_in[4];
    const float* b2 = (const float*)d_in[5];

    const int N = in_sizes[0] / 128;  // 100000
    const int E = in_sizes[1] / 2;    // 1600000
    const int* src = ei;
    const int* dst = ei + E;

    // Workspace: dinv | h1 (reused as g2) | agg1  (~103 MB)
    float* ws   = (float*)d_ws;
    size_t off  = 0;
    float* dinv = ws + off; off += ((size_t)N + 255) & ~(size_t)255;
    float* h1   = ws + off; off += (size_t)N * 128;
    float* agg1 = ws + off; off += (size_t)N * 128;
    float* g2   = h1;  // layer-1 h no longer needed once agg1 is final
    float* out  = (float*)d_out;

    const int tpb = 256;
    // degree -> dinv
    zero_f32 <<<(N + tpb - 1) / tpb, tpb, 0, stream>>>(dinv, N);
    count_deg<<<(E + tpb - 1) / tpb, tpb, 0, stream>>>(dst, dinv, E);
    make_dinv<<<(N + tpb - 1) / tpb, tpb, 0, stream>>>(dinv, N);

    // Layer 1: h1 = x @ W1 ; agg1 = h1*dinv^2 + b1 ; scatter edges
    gcn_gemm<128, false><<<(N + 15) / 16, 256, 0, stream>>>(x, W1, b1, dinv, h1, agg1, N);
    gcn_scatter<128><<<(E + 7) / 8, 256, 0, stream>>>(h1, dinv, src, dst, agg1, E);

    // Layer 2: g2 = relu(agg1) @ W2 ; out = g2*dinv^2 + b2 ; scatter edges
    gcn_gemm<64, true><<<(N + 31) / 32, 256, 0, stream>>>(agg1, W2, b2, dinv, g2, out, N);
    gcn_scatter<64><<<(E + 7) / 8, 256, 0, stream>>>(g2, dinv, src, dst, out, E);
}